// Detection_CrossEntropy_37391985279242
// MI455X (gfx1250) — compile-verified
//
#include <hip/hip_runtime.h>

typedef float v2f __attribute__((ext_vector_type(2)));
typedef float v8f __attribute__((ext_vector_type(8)));

#define BB 8
#define NN 25200
#define GG 32
#define CC 80
#define ROW 85
#define MWF 640.0f
#define TILES (NN / 16)          // 1575 tiles of 16 rows
#define WPB 8                    // waves per block (256 threads)

// ws layout (floats): [0..7] = num[b], [8..15] = den[b]
__global__ void det_ce_zero(float* __restrict__ ws) {
    if (threadIdx.x < 16) ws[threadIdx.x] = 0.0f;
}

__global__ void det_ce_main(const float* __restrict__ outp,
                            const float* __restrict__ label,
                            const int* __restrict__ thr_p,
                            float* __restrict__ ws) {
    const int lane = threadIdx.x & 31;
    const int wid  = threadIdx.x >> 5;
    const int b    = blockIdx.y;
    const int tile = blockIdx.x * WPB + wid;

    __shared__ float s_gt[GG * 4];
    __shared__ int   s_cls[GG];
    __shared__ float s_num[WPB], s_den[WPB];

    // Per-block: build GT boxes (clip(xywh2xyxy,0,1)*640) and class ids in LDS
    if (threadIdx.x < GG) {
        const float* L = label + ((size_t)b * GG + threadIdx.x) * 5;
        float cx = L[1], cy = L[2], w = L[3], h = L[4];
        float x1 = cx - 0.5f * w, y1 = cy - 0.5f * h;
        float x2 = cx + 0.5f * w, y2 = cy + 0.5f * h;
        x1 = fminf(fmaxf(x1, 0.0f), 1.0f) * MWF;
        y1 = fminf(fmaxf(y1, 0.0f), 1.0f) * MWF;
        x2 = fminf(fmaxf(x2, 0.0f), 1.0f) * MWF;
        y2 = fminf(fmaxf(y2, 0.0f), 1.0f) * MWF;
        s_gt[threadIdx.x * 4 + 0] = x1;
        s_gt[threadIdx.x * 4 + 1] = y1;
        s_gt[threadIdx.x * 4 + 2] = x2;
        s_gt[threadIdx.x * 4 + 3] = y2;
        s_cls[threadIdx.x] = (int)L[0];
    }
    __syncthreads();

    float num = 0.0f, den = 0.0f;

    if (tile < TILES) {                    // wave-uniform guard: EXEC all-1s inside
        const int r = lane & 15;           // row of the 16-row tile (M index)
        const int h = lane >> 4;           // wave half selects K slots {0,1} vs {2,3}
        const int n = tile * 16 + r;
        const float* row = outp + ((size_t)b * NN + n) * ROW;

        const float obj = row[4];

        // Gather this lane's 40 scaled logits (classes 4j+2h, 4j+2h+1), track max
        float2 pl[20];
        float mx = -3.402823466e38f;
        #pragma unroll
        for (int j = 0; j < 20; ++j) {
            const float* p = row + 5 + 4 * j + 2 * h;
            float a0 = p[0] * obj;
            float a1 = p[1] * obj;
            pl[j].x = a0; pl[j].y = a1;
            mx = fmaxf(mx, fmaxf(a0, a1));
        }
        // Combine the two half-row maxima -> full 80-class max per row
        mx = fmaxf(mx, __shfl_xor(mx, 16, 32));

        // Sum of exp over 80 classes via 20 chained f32 WMMAs with ones-B:
        // D[m,j] = sum_k A[m,k]*1 + C[m,j]  (row-sum accumulate; exact fp32)
        v2f ones; ones.x = 1.0f; ones.y = 1.0f;
        v8f acc = {};
        #pragma unroll
        for (int j = 0; j < 20; ++j) {
            v2f a;
            a.x = __expf(pl[j].x - mx);
            a.y = __expf(pl[j].y - mx);
            acc = __builtin_amdgcn_wmma_f32_16x16x4_f32(
                false, a, false, ones, (short)0, acc, false, false);
        }
        // Extract row r's sum: rows 0-7 live in lanes 0-15 reg[row],
        // rows 8-15 live in lanes 16-31 reg[row-8].
        int i = r & 7;
        float z = (i < 4) ? ((i < 2) ? ((i == 0) ? acc[0] : acc[1])
                                     : ((i == 2) ? acc[2] : acc[3]))
                          : ((i < 6) ? ((i == 4) ? acc[4] : acc[5])
                                     : ((i == 6) ? acc[6] : acc[7]));
        int srcLane = (r < 8) ? r : (r + 8);
        float sumexp = __shfl(z, srcLane, 32);
        float lse = mx + __logf(sumexp);

        // Prediction box -> xyxy, area
        float px = row[0], py = row[1], pw = row[2], ph = row[3];
        float px1 = px - 0.5f * pw, py1 = py - 0.5f * ph;
        float px2 = px + 0.5f * pw, py2 = py + 0.5f * ph;
        float parea = (px2 - px1) * (py2 - py1);
        const float thr = (float)thr_p[0];
        const bool keep = obj >= thr;

        // This lane handles 16 of the 32 GTs for its row
        #pragma unroll
        for (int gg = 0; gg < 16; ++gg) {
            int g = h * 16 + gg;
            float gx1 = s_gt[g * 4 + 0], gy1 = s_gt[g * 4 + 1];
            float gx2 = s_gt[g * 4 + 2], gy2 = s_gt[g * 4 + 3];
            float garea = (gx2 - gx1) * (gy2 - gy1);
            float ix = fmaxf(fminf(gx2, px2) - fmaxf(gx1, px1), 0.0f);
            float iy = fmaxf(fminf(gy2, py2) - fmaxf(gy1, py1), 0.0f);
            float inter = ix * iy;
            float iou = inter / (garea + parea - inter);
            if ((iou >= 0.5f) && keep) {
                float plc = row[5 + s_cls[g]] * obj;
                num += lse - plc;
                den += 1.0f;
            }
        }
    }

    // Wave reduction (wave32)
    #pragma unroll
    for (int off = 16; off > 0; off >>= 1) {
        num += __shfl_xor(num, off, 32);
        den += __shfl_xor(den, off, 32);
    }
    if (lane == 0) { s_num[wid] = num; s_den[wid] = den; }
    __syncthreads();
    if (threadIdx.x == 0) {
        float tn = 0.0f, td = 0.0f;
        #pragma unroll
        for (int w = 0; w < WPB; ++w) { tn += s_num[w]; td += s_den[w]; }
        atomicAdd(&ws[b], tn);
        atomicAdd(&ws[8 + b], td);
    }
}

__global__ void det_ce_finish(const float* __restrict__ ws, float* __restrict__ out) {
    if (threadIdx.x < BB) out[threadIdx.x] = ws[threadIdx.x] / ws[8 + threadIdx.x];
}

extern "C" void kernel_launch(void* const* d_in, const int* in_sizes, int n_in,
                              void* d_out, int out_size, void* d_ws, size_t ws_size,
                              hipStream_t stream) {
    const float* outp  = (const float*)d_in[0];   // [B, N, 85] fp32
    const float* label = (const float*)d_in[1];   // [B, G, 5]  fp32
    const int*   thr   = (const int*)d_in[2];     // scalar int
    float* ws  = (float*)d_ws;
    float* out = (float*)d_out;                   // [1, B] fp32

    det_ce_zero<<<1, 32, 0, stream>>>(ws);
    dim3 grid((TILES + WPB - 1) / WPB, BB);       // (197, 8)
    det_ce_main<<<grid, 256, 0, stream>>>(outp, label, thr, ws);
    det_ce_finish<<<1, 32, 0, stream>>>(ws, out);
}